// PromptMemory_11802570130390
// MI455X (gfx1250) — compile-verified
//
#include <hip/hip_runtime.h>
#include <math.h>

// ---------------------------------------------------------------------------
// Problem constants (match reference)
// ---------------------------------------------------------------------------
#define B_   512
#define E_   4096
#define K_   1024
#define M_   1024
#define L_   32
#define TOPK 4
#define EPSN 1e-12f

typedef __attribute__((ext_vector_type(2))) float v2f;
typedef __attribute__((ext_vector_type(4))) float v4f;
typedef __attribute__((ext_vector_type(8))) float v8f;

// ---------------------------------------------------------------------------
// NT GEMM via V_WMMA_F32_16X16X4_F32:  C[Mr x N] = A[Mr x Kd] * B[N x Kd]^T
//
// One wave32 computes a 16x64 strip: 4 accumulators (16x16 each), A fragment
// reused across the 4 column tiles -> 5 b128 loads per 8 WMMAs.
//
// K-permutation trick: summation over Kd is order independent, so for each
// group of 8 K-values, half-wave 0 loads float4 at {e..e+3} and half-wave 1
// at {e+4..e+7}.  WMMA #0 consumes components (x,y) => K-set
// {e,e+1,e+4,e+5}; WMMA #1 consumes (z,w) => K-set {e+2,e+3,e+6,e+7}.
// A and B use the identical lane->K mapping, so each product pairs matching
// K's and every K in [e, e+8) is covered exactly once.
//
// fp32 A-fragment (16x4): lanes 0-15 hold M=lane, K={k0,k0+1}; lanes 16-31
// hold M=lane-16, K={k0+2,k0+3}.  B mirrors with N in lane&15.
// C/D layout: VGPR r -> row (half*8 + r), col = lane&15.
// ---------------------------------------------------------------------------
__global__ __launch_bounds__(32)
void gemm_nt_wmma_f32(const float* __restrict__ A,
                      const float* __restrict__ Bm,
                      float* __restrict__ C,
                      int Kd, int N) {
    const int lane = threadIdx.x;        // full wave, EXEC all-1s (WMMA req)
    const int half = lane >> 4;
    const int l16  = lane & 15;
    const int rowBase = blockIdx.y * 16; // output rows
    const int colBase = blockIdx.x * 64; // output cols (4 x 16 tiles)

    const float* arow = A  + (size_t)(rowBase + l16) * Kd + 4 * half;
    const float* b0   = Bm + (size_t)(colBase + l16) * Kd + 4 * half;
    const float* b1   = b0 + (size_t)16 * Kd;
    const float* b2   = b0 + (size_t)32 * Kd;
    const float* b3   = b0 + (size_t)48 * Kd;

    v8f acc0 = {}, acc1 = {}, acc2 = {}, acc3 = {};

#pragma unroll 2
    for (int e = 0; e < Kd; e += 8) {
        v4f af  = *(const v4f*)(arow + e);
        v4f bf0 = *(const v4f*)(b0 + e);
        v4f bf1 = *(const v4f*)(b1 + e);
        v4f bf2 = *(const v4f*)(b2 + e);
        v4f bf3 = *(const v4f*)(b3 + e);

        v2f aL = {af.x, af.y},  aH = {af.z, af.w};
        v2f l0 = {bf0.x, bf0.y}, h0 = {bf0.z, bf0.w};
        v2f l1 = {bf1.x, bf1.y}, h1 = {bf1.z, bf1.w};
        v2f l2 = {bf2.x, bf2.y}, h2 = {bf2.z, bf2.w};
        v2f l3 = {bf3.x, bf3.y}, h3 = {bf3.z, bf3.w};

        acc0 = __builtin_amdgcn_wmma_f32_16x16x4_f32(false, aL, false, l0, (short)0, acc0, false, false);
        acc1 = __builtin_amdgcn_wmma_f32_16x16x4_f32(false, aL, false, l1, (short)0, acc1, false, false);
        acc2 = __builtin_amdgcn_wmma_f32_16x16x4_f32(false, aL, false, l2, (short)0, acc2, false, false);
        acc3 = __builtin_amdgcn_wmma_f32_16x16x4_f32(false, aL, false, l3, (short)0, acc3, false, false);
        acc0 = __builtin_amdgcn_wmma_f32_16x16x4_f32(false, aH, false, h0, (short)0, acc0, false, false);
        acc1 = __builtin_amdgcn_wmma_f32_16x16x4_f32(false, aH, false, h1, (short)0, acc1, false, false);
        acc2 = __builtin_amdgcn_wmma_f32_16x16x4_f32(false, aH, false, h2, (short)0, acc2, false, false);
        acc3 = __builtin_amdgcn_wmma_f32_16x16x4_f32(false, aH, false, h3, (short)0, acc3, false, false);
    }

    float* cbase = C + (size_t)(rowBase + half * 8) * N + l16;
#pragma unroll
    for (int r = 0; r < 8; ++r) {
        float* crow = cbase + (size_t)r * N;
        crow[colBase +  0] = acc0[r];
        crow[colBase + 16] = acc1[r];
        crow[colBase + 32] = acc2[r];
        crow[colBase + 48] = acc3[r];
    }
}

// ---------------------------------------------------------------------------
// Per-row inverse L2 norm:  inv[row] = 1 / max(||X[row,:]||, eps)
// One 256-thread block per row.
// ---------------------------------------------------------------------------
__global__ __launch_bounds__(256)
void rownorm_inv(const float* __restrict__ X, float* __restrict__ inv, int rowLen) {
    __shared__ float red[256];
    const float* row = X + (size_t)blockIdx.x * rowLen;
    float s = 0.f;
    for (int i = threadIdx.x; i < rowLen; i += 256) {
        float v = row[i];
        s += v * v;
    }
    red[threadIdx.x] = s;
    __syncthreads();
#pragma unroll
    for (int off = 128; off > 0; off >>= 1) {
        if (threadIdx.x < off) red[threadIdx.x] += red[threadIdx.x + off];
        __syncthreads();
    }
    if (threadIdx.x == 0) {
        float n = sqrtf(red[0]);
        inv[blockIdx.x] = 1.f / fmaxf(n, EPSN);
    }
}

// ---------------------------------------------------------------------------
// Top-4 + softmax per batch row.
// sim_raw[b,m] = x_proj[b] . keys[m]; true similarity is
// sim_raw * invx[b] * invk[m].  invx[b] > 0 is a per-row constant, so the
// ranking only needs sim_raw * invk[m]; invx[b] is applied before softmax.
// One wave32 per row: per-lane sorted top-4, LDS merge of 128 candidates.
// ---------------------------------------------------------------------------
__global__ __launch_bounds__(32)
void topk_softmax(const float* __restrict__ sim,
                  const float* __restrict__ invx,
                  const float* __restrict__ invk,
                  float* __restrict__ wout,
                  int* __restrict__ iout) {
    __shared__ float sv[32 * TOPK];
    __shared__ int   si[32 * TOPK];

    const int b    = blockIdx.x;
    const int lane = threadIdx.x;
    const float* srow = sim + (size_t)b * M_;

    float v0 = -1e30f, v1 = -1e30f, v2 = -1e30f, v3 = -1e30f;
    int   i0 = 0, i1 = 0, i2 = 0, i3 = 0;
    for (int m = lane; m < M_; m += 32) {
        float v = srow[m] * invk[m];
        if (v > v3) {
            if (v > v0)      { v3=v2;i3=i2; v2=v1;i2=i1; v1=v0;i1=i0; v0=v;i0=m; }
            else if (v > v1) { v3=v2;i3=i2; v2=v1;i2=i1; v1=v; i1=m; }
            else if (v > v2) { v3=v2;i3=i2; v2=v; i2=m; }
            else             { v3=v; i3=m; }
        }
    }
    sv[lane*4+0] = v0; si[lane*4+0] = i0;
    sv[lane*4+1] = v1; si[lane*4+1] = i1;
    sv[lane*4+2] = v2; si[lane*4+2] = i2;
    sv[lane*4+3] = v3; si[lane*4+3] = i3;
    __syncthreads();

    if (lane == 0) {
        float tv0=-1e30f, tv1=-1e30f, tv2=-1e30f, tv3=-1e30f;
        int   ti0=0, ti1=0, ti2=0, ti3=0;
        for (int c = 0; c < 32 * TOPK; ++c) {
            float v = sv[c];
            int   m = si[c];
            if (v > tv3) {
                if (v > tv0)      { tv3=tv2;ti3=ti2; tv2=tv1;ti2=ti1; tv1=tv0;ti1=ti0; tv0=v;ti0=m; }
                else if (v > tv1) { tv3=tv2;ti3=ti2; tv2=tv1;ti2=ti1; tv1=v; ti1=m; }
                else if (v > tv2) { tv3=tv2;ti3=ti2; tv2=v; ti2=m; }
                else              { tv3=v; ti3=m; }
            }
        }
        const float sx = invx[b];
        float r0 = tv0 * sx, r1 = tv1 * sx, r2 = tv2 * sx, r3 = tv3 * sx;
        float mx = r0;  // tv0 is the max
        float e0 = __expf(r0 - mx), e1 = __expf(r1 - mx),
              e2 = __expf(r2 - mx), e3 = __expf(r3 - mx);
        float den = 1.f / (e0 + e1 + e2 + e3);
        wout[b*TOPK+0] = e0 * den;  iout[b*TOPK+0] = ti0;
        wout[b*TOPK+1] = e1 * den;  iout[b*TOPK+1] = ti1;
        wout[b*TOPK+2] = e2 * den;  iout[b*TOPK+2] = ti2;
        wout[b*TOPK+3] = e3 * den;  iout[b*TOPK+3] = ti3;
    }
}

// ---------------------------------------------------------------------------
// out[b,l,:] = sum_t w[b,t] * PM[idx[b,t], l, :]
// One 256-thread block per (b,l); 128-bit streaming.  This stage is the HBM
// roofline (~1 GB read + 256 MB written).  Output is write-once, so use
// non-temporal stores (TH=NT) to keep the 192 MB L2 free for prompt_memory
// reuse across batch rows.
// ---------------------------------------------------------------------------
__global__ __launch_bounds__(256)
void gather_wsum(const float* __restrict__ PM,
                 const float* __restrict__ w,
                 const int* __restrict__ idx,
                 float* __restrict__ out) {
    const int bl = blockIdx.x;          // b*L + l
    const int b  = bl / L_;
    const int l  = bl - b * L_;

    const float w0 = w[b*TOPK+0], w1 = w[b*TOPK+1],
                w2 = w[b*TOPK+2], w3 = w[b*TOPK+3];
    const v4f* p0 = (const v4f*)(PM + ((size_t)idx[b*TOPK+0] * L_ + l) * E_);
    const v4f* p1 = (const v4f*)(PM + ((size_t)idx[b*TOPK+1] * L_ + l) * E_);
    const v4f* p2 = (const v4f*)(PM + ((size_t)idx[b*TOPK+2] * L_ + l) * E_);
    const v4f* p3 = (const v4f*)(PM + ((size_t)idx[b*TOPK+3] * L_ + l) * E_);
    v4f* o = (v4f*)(out + (size_t)bl * E_);

#pragma unroll
    for (int e = threadIdx.x; e < E_ / 4; e += 256) {
        v4f a0 = p0[e], a1 = p1[e], a2 = p2[e], a3 = p3[e];
        v4f r = w0 * a0 + w1 * a1 + w2 * a2 + w3 * a3;
        __builtin_nontemporal_store(r, o + e);   // keep L2 for PM reuse
    }
}

// ---------------------------------------------------------------------------
// Launch
// ---------------------------------------------------------------------------
extern "C" void kernel_launch(void* const* d_in, const int* in_sizes, int n_in,
                              void* d_out, int out_size, void* d_ws, size_t ws_size,
                              hipStream_t stream) {
    const float* x    = (const float*)d_in[0];   // [B,E]
    const float* W    = (const float*)d_in[1];   // [K,E]
    const float* PM   = (const float*)d_in[2];   // [M,L,E]
    const float* keys = (const float*)d_in[3];   // [M,K]
    float* out = (float*)d_out;                  // [B,L,E]

    // Workspace carve-up (~4.2 MB total)
    float* xproj = (float*)d_ws;                    // B*K
    float* sim   = xproj + (size_t)B_ * K_;         // B*M
    float* invx  = sim   + (size_t)B_ * M_;         // B
    float* invk  = invx  + B_;                      // M
    float* wbuf  = invk  + M_;                      // B*TOPK
    int*   ibuf  = (int*)(wbuf + (size_t)B_ * TOPK);// B*TOPK

    // 1) x_proj = X @ W^T          [512 x 1024], Kdim = 4096
    {
        dim3 grid(K_ / 64, B_ / 16);
        gemm_nt_wmma_f32<<<grid, 32, 0, stream>>>(x, W, xproj, E_, K_);
    }
    // 2) inverse row norms
    rownorm_inv<<<B_, 256, 0, stream>>>(xproj, invx, K_);
    rownorm_inv<<<M_, 256, 0, stream>>>(keys,  invk, K_);
    // 3) sim_raw = x_proj @ keys^T [512 x 1024], Kdim = 1024
    {
        dim3 grid(M_ / 64, B_ / 16);
        gemm_nt_wmma_f32<<<grid, 32, 0, stream>>>(xproj, keys, sim, K_, M_);
    }
    // 4) top-4 + softmax weights
    topk_softmax<<<B_, 32, 0, stream>>>(sim, invx, invk, wbuf, ibuf);
    // 5) softmax-weighted gather-sum (bandwidth-dominant stage)
    gather_wsum<<<B_ * L_, 256, 0, stream>>>(PM, wbuf, ibuf, out);
}